// DNC_48335561949968
// MI455X (gfx1250) — compile-verified
//
#include <hip/hip_runtime.h>
#include <hip/hip_bf16.h>
#include <math.h>

// DNC forward for MI455X (gfx1250, wave32).
// B=32, T=64, H=512, M=128, R=4, W=64.
#define B_   32
#define T_   64
#define H_   512
#define M_   128
#define R_   4
#define W_   64
#define ISZ  768      // R*W + H
#define XI_N 471
#define XI_PAD 480
#define G4H  2048     // 4*H
#define EPS_ 1e-6f

typedef __attribute__((ext_vector_type(2))) float v2f;
typedef __attribute__((ext_vector_type(8))) float v8f;

// ---- workspace layout (float offsets) ----
#define OFF_H       0
#define OFF_C       (OFF_H     + B_*H_)
#define OFF_MEM     (OFF_C     + B_*H_)
#define OFF_LINK    (OFF_MEM   + B_*M_*W_)
#define OFF_PREC    (OFF_LINK  + B_*M_*M_)
#define OFF_RW      (OFF_PREC  + B_*M_)
#define OFF_WW      (OFF_RW    + B_*R_*M_)
#define OFF_USAGE   (OFF_WW    + B_*M_)
#define OFF_LREAD   (OFF_USAGE + B_*M_)
#define STATE_TOTAL (OFF_LREAD + B_*R_*W_)
#define OFF_GATES   STATE_TOTAL
#define OFF_XI      (OFF_GATES + B_*G4H)

__device__ __forceinline__ float sigm_(float x) { return 1.0f / (1.0f + expf(-x)); }
__device__ __forceinline__ float softplus_(float x) {
  return fmaxf(x, 0.0f) + log1pf(expf(-fabsf(x)));
}
__device__ __forceinline__ v2f ld2_(const float* p) { return *(const v2f*)p; }

// One K-segment of the f32 WMMA GEMM. Per lane: half=lane>>4, so the pointer
// already includes +2*half; contiguous float2 covers this lane's two K values.
// Two M tiles (rows 0-15 and 16-31) share the B fragment.
__device__ __forceinline__ void wmma_seg(const float* a0p, const float* a1p,
                                         const float* bp, int K,
                                         v8f& acc0, v8f& acc1) {
  for (int k = 0; k < K; k += 4) {
    v2f a0 = ld2_(a0p + k);
    v2f a1 = ld2_(a1p + k);
    v2f bb = ld2_(bp + k);
    acc0 = __builtin_amdgcn_wmma_f32_16x16x4_f32(false, a0, false, bb,
                                                 (short)0, acc0, false, false);
    acc1 = __builtin_amdgcn_wmma_f32_16x16x4_f32(false, a1, false, bb,
                                                 (short)0, acc1, false, false);
  }
}

// ---- init: zero state, mem = EPS ----
__global__ void init_state(float* ws) {
  int idx = blockIdx.x * blockDim.x + threadIdx.x;
  int stride = gridDim.x * blockDim.x;
  for (int i = idx; i < STATE_TOTAL; i += stride) {
    ws[i] = (i >= OFF_MEM && i < OFF_MEM + B_*M_*W_) ? EPS_ : 0.0f;
  }
}

// ---- gates = [x_t | last_read] @ W_ih^T + h @ W_hh^T + b_ih + b_hh ----
// grid 16 x block 256 (8 waves): 128 N-tiles of 16; each wave does both M tiles.
__global__ __launch_bounds__(256) void gates_gemm(
    const float* __restrict__ x, const float* __restrict__ lread,
    const float* __restrict__ h, const float* __restrict__ W_ih,
    const float* __restrict__ W_hh, const float* __restrict__ b_ih,
    const float* __restrict__ b_hh, float* __restrict__ gates, int t) {
  int lane = threadIdx.x & 31;
  int tile = blockIdx.x * (blockDim.x >> 5) + (threadIdx.x >> 5);
  int half = lane >> 4, lid = lane & 15;
  int n0 = tile * 16;
  int j = n0 + lid;                       // output column / B-matrix column
  v8f acc0 = {0,0,0,0,0,0,0,0};
  v8f acc1 = {0,0,0,0,0,0,0,0};
  const int ko = 2 * half;
  // segment 1: x_t (K = 512)
  wmma_seg(x + ((size_t)lid      * T_ + t) * H_ + ko,
           x + ((size_t)(lid+16) * T_ + t) * H_ + ko,
           W_ih + (size_t)j * ISZ + ko, H_, acc0, acc1);
  // segment 2: last_read (K = 256)
  wmma_seg(lread + lid * (R_*W_) + ko,
           lread + (lid+16) * (R_*W_) + ko,
           W_ih + (size_t)j * ISZ + H_ + ko, R_*W_, acc0, acc1);
  // segment 3: h @ W_hh^T (K = 512)
  wmma_seg(h + lid * H_ + ko, h + (lid+16) * H_ + ko,
           W_hh + (size_t)j * H_ + ko, H_, acc0, acc1);
  float bias = b_ih[j] + b_hh[j];
  #pragma unroll
  for (int e = 0; e < 8; ++e) {
    gates[(e + 8*half)      * G4H + j] = acc0[e] + bias;
    gates[(e + 8*half + 16) * G4H + j] = acc1[e] + bias;
  }
}

// ---- LSTM pointwise cell ----
__global__ void lstm_cell(const float* __restrict__ gates, float* h, float* c) {
  int idx = blockIdx.x * blockDim.x + threadIdx.x;
  if (idx >= B_ * H_) return;
  int b = idx / H_, jj = idx - b * H_;
  const float* g = gates + b * G4H;
  float ig = sigm_(g[jj]);
  float fg = sigm_(g[H_ + jj]);
  float gg = tanhf(g[2*H_ + jj]);
  float og = sigm_(g[3*H_ + jj]);
  float cn = fg * c[idx] + ig * gg;
  c[idx] = cn;
  h[idx] = og * tanhf(cn);
}

// ---- xi = h @ W_xi^T + b_xi (N=471 padded to 480) ----
// grid 15 x block 64 (2 waves): 30 N-tiles.
__global__ __launch_bounds__(64) void xi_gemm(
    const float* __restrict__ h, const float* __restrict__ W_xi,
    const float* __restrict__ b_xi, float* __restrict__ xiw) {
  int lane = threadIdx.x & 31;
  int tile = blockIdx.x * (blockDim.x >> 5) + (threadIdx.x >> 5);
  int half = lane >> 4, lid = lane & 15;
  int j = tile * 16 + lid;
  int jc = j < XI_N ? j : (XI_N - 1);     // clamp weight row for OOB lanes
  v8f acc0 = {0,0,0,0,0,0,0,0};
  v8f acc1 = {0,0,0,0,0,0,0,0};
  const int ko = 2 * half;
  wmma_seg(h + lid * H_ + ko, h + (lid+16) * H_ + ko,
           W_xi + (size_t)jc * H_ + ko, H_, acc0, acc1);
  if (j < XI_N) {
    float bias = b_xi[jc];
    #pragma unroll
    for (int e = 0; e < 8; ++e) {
      xiw[(e + 8*half)      * XI_PAD + j] = acc0[e] + bias;
      xiw[(e + 8*half + 16) * XI_PAD + j] = acc1[e] + bias;
    }
  }
}

// ---- out[:,t,:] = [h | last_read] @ W_o^T + b_o ----
// grid 4 x block 256 (8 waves): 32 N-tiles.
__global__ __launch_bounds__(256) void out_gemm(
    const float* __restrict__ h, const float* __restrict__ lread,
    const float* __restrict__ W_o, const float* __restrict__ b_o,
    float* __restrict__ out, int t) {
  int lane = threadIdx.x & 31;
  int tile = blockIdx.x * (blockDim.x >> 5) + (threadIdx.x >> 5);
  int half = lane >> 4, lid = lane & 15;
  int j = tile * 16 + lid;
  v8f acc0 = {0,0,0,0,0,0,0,0};
  v8f acc1 = {0,0,0,0,0,0,0,0};
  const int ko = 2 * half;
  wmma_seg(h + lid * H_ + ko, h + (lid+16) * H_ + ko,
           W_o + (size_t)j * ISZ + ko, H_, acc0, acc1);
  wmma_seg(lread + lid * (R_*W_) + ko, lread + (lid+16) * (R_*W_) + ko,
           W_o + (size_t)j * ISZ + H_ + ko, R_*W_, acc0, acc1);
  float bias = b_o[j];
  #pragma unroll
  for (int e = 0; e < 8; ++e) {
    int r0 = e + 8*half, r1 = r0 + 16;
    out[((size_t)r0 * T_ + t) * H_ + j] = acc0[e] + bias;
    out[((size_t)r1 * T_ + t) * H_ + j] = acc1[e] + bias;
  }
}

// ---- per-batch DNC memory machinery: one block of 128 threads per batch ----
__device__ __forceinline__ float red_sum_(float v, float* s, int tid) {
  s[tid] = v; __syncthreads();
  #pragma unroll
  for (int o = 64; o > 0; o >>= 1) {
    if (tid < o) s[tid] += s[tid + o];
    __syncthreads();
  }
  float r = s[0]; __syncthreads(); return r;
}
__device__ __forceinline__ float red_max_(float v, float* s, int tid) {
  s[tid] = v; __syncthreads();
  #pragma unroll
  for (int o = 64; o > 0; o >>= 1) {
    if (tid < o) s[tid] = fmaxf(s[tid], s[tid + o]);
    __syncthreads();
  }
  float r = s[0]; __syncthreads(); return r;
}

__global__ __launch_bounds__(128) void memory_step(
    float* __restrict__ mem, float* __restrict__ link, float* __restrict__ prec,
    float* __restrict__ rw, float* __restrict__ ww, float* __restrict__ usage,
    float* __restrict__ lread, const float* __restrict__ xi) {
  __shared__ float s_mem[M_][W_];                 // 32 KB: new memory tile
  __shared__ float s_wkey[W_], s_erase[W_], s_wvec[W_];
  __shared__ float s_rkeys[R_][W_];
  __shared__ float s_rstr[R_], s_fgate[R_], s_modes[R_][3];
  __shared__ float s_scal[3];                     // write_str, alloc_g, write_g
  __shared__ float s_wwold[M_], s_prec[M_], s_wwnew[M_];
  __shared__ float s_rwold[R_][M_], s_rwnew[R_][M_];
  __shared__ float s_ubuf[M_], s_sorted[M_], s_shifted[M_];
  __shared__ float s_fwd[R_][M_], s_bwd[R_][M_];
  __shared__ float s_red[M_];

  const int b = blockIdx.x, tid = threadIdx.x;
  const float* xb = xi + b * XI_PAD;

  // ---- load + activate interface vector, load state ----
  if (tid < W_) {
    s_wkey[tid]  = tanhf(xb[260 + tid]);
    s_erase[tid] = sigm_(xb[325 + tid]);
    s_wvec[tid]  = tanhf(xb[389 + tid]);
    #pragma unroll
    for (int r = 0; r < R_; ++r) s_rkeys[r][tid] = tanhf(xb[r * W_ + tid]);
  }
  if (tid < R_) {
    s_rstr[tid]  = softplus_(xb[256 + tid]);
    s_fgate[tid] = sigm_(xb[453 + tid]);
    float m0 = xb[459 + tid*3], m1 = xb[460 + tid*3], m2 = xb[461 + tid*3];
    float mx = fmaxf(m0, fmaxf(m1, m2));
    float e0 = expf(m0 - mx), e1 = expf(m1 - mx), e2 = expf(m2 - mx);
    float es = e0 + e1 + e2;
    s_modes[tid][0] = e0/es; s_modes[tid][1] = e1/es; s_modes[tid][2] = e2/es;
  }
  if (tid == 0) {
    s_scal[0] = softplus_(xb[324]);  // write_str
    s_scal[1] = sigm_(xb[457]);      // alloc_g
    s_scal[2] = sigm_(xb[458]);      // write_g
  }
  s_wwold[tid] = ww[b*M_ + tid];
  s_prec[tid]  = prec[b*M_ + tid];
  #pragma unroll
  for (int r = 0; r < R_; ++r) s_rwold[r][tid] = rw[(b*R_ + r)*M_ + tid];
  float u = usage[b*M_ + tid];
  __syncthreads();

  // ---- usage update (uses previous ww / rw) ----
  u = u + (1.0f - u) * s_wwold[tid];
  float psi = 1.0f;
  #pragma unroll
  for (int r = 0; r < R_; ++r) psi *= (1.0f - s_fgate[r] * s_rwold[r][tid]);
  u *= psi;
  usage[b*M_ + tid] = u;

  // ---- write content weighting on OLD memory ----
  const float* mrow = mem + ((size_t)b*M_ + tid) * W_;
  float dot = 0.0f, nm2 = 0.0f, nk2 = 0.0f;
  for (int w = 0; w < W_; ++w) {
    float mv = mrow[w], kv = s_wkey[w];
    dot += mv * kv; nm2 += mv * mv; nk2 += kv * kv;
  }
  float score = dot / (sqrtf(nk2) * sqrtf(nm2) + EPS_) * s_scal[0];
  float mx = red_max_(score, s_red, tid);
  float ex = expf(score - mx);
  float wcw = ex / red_sum_(ex, s_red, tid);

  // ---- allocation: stable rank sort of usage ----
  float uu = 5e-6f + (1.0f - 5e-6f) * u;
  s_ubuf[tid] = uu; __syncthreads();
  int rank = 0;
  for (int jm = 0; jm < M_; ++jm) {
    float uj = s_ubuf[jm];
    rank += (uj < uu) || (uj == uu && jm < tid);
  }
  s_sorted[rank] = uu; __syncthreads();
  if (tid == 0) {
    float p = 1.0f;
    for (int q = 0; q < M_; ++q) { s_shifted[q] = p; p *= s_sorted[q]; }
  }
  __syncthreads();
  float alloc = (1.0f - uu) * s_shifted[rank];

  // ---- write weighting ----
  float ag = s_scal[1], wg = s_scal[2];
  float wwn = wg * (ag * alloc + (1.0f - ag) * wcw);
  s_wwnew[tid] = wwn;

  // ---- erase + write memory (into LDS tile + global) ----
  float* mroww = mem + ((size_t)b*M_ + tid) * W_;
  for (int w = 0; w < W_; ++w) {
    float mv = mrow[w] * (1.0f - wwn * s_erase[w]) + wwn * s_wvec[w];
    s_mem[tid][w] = mv;
    mroww[w] = mv;
  }
  __syncthreads();

  float wwsum = red_sum_(wwn, s_red, tid);

  // ---- link update (row pass) fused with fwd = link_new @ rw_old ----
  float* lrow = link + ((size_t)b*M_ + tid) * M_;
  float f0 = 0, f1 = 0, f2 = 0, f3 = 0;
  for (int jm = 0; jm < M_; ++jm) {
    float l = lrow[jm];
    l = (1.0f - wwn - s_wwnew[jm]) * l + wwn * s_prec[jm];
    if (jm == tid) l = 0.0f;
    lrow[jm] = l;
    f0 += l * s_rwold[0][jm]; f1 += l * s_rwold[1][jm];
    f2 += l * s_rwold[2][jm]; f3 += l * s_rwold[3][jm];
  }
  s_fwd[0][tid] = f0; s_fwd[1][tid] = f1; s_fwd[2][tid] = f2; s_fwd[3][tid] = f3;
  prec[b*M_ + tid] = (1.0f - wwsum) * s_prec[tid] + wwn;
  ww[b*M_ + tid] = wwn;
  __syncthreads();

  // ---- bwd[r][j] = sum_i rw_old[r][i] * link_new[i][j] (column pass) ----
  const float* lbase = link + (size_t)b*M_*M_;
  float g0 = 0, g1 = 0, g2 = 0, g3 = 0;
  for (int im = 0; im < M_; ++im) {
    float l = lbase[(size_t)im*M_ + tid];
    g0 += s_rwold[0][im] * l; g1 += s_rwold[1][im] * l;
    g2 += s_rwold[2][im] * l; g3 += s_rwold[3][im] * l;
  }
  s_bwd[0][tid] = g0; s_bwd[1][tid] = g1; s_bwd[2][tid] = g2; s_bwd[3][tid] = g3;

  // ---- content read weighting on NEW memory + mode mix ----
  for (int r = 0; r < R_; ++r) {
    float d = 0, m2 = 0, k2 = 0;
    for (int w = 0; w < W_; ++w) {
      float mv = s_mem[tid][w], kv = s_rkeys[r][w];
      d += mv * kv; m2 += mv * mv; k2 += kv * kv;
    }
    float sc = d / (sqrtf(k2) * sqrtf(m2) + EPS_) * s_rstr[r];
    float rmx = red_max_(sc, s_red, tid);
    float rex = expf(sc - rmx);
    float ccw = rex / red_sum_(rex, s_red, tid);
    float rwn = s_modes[r][0] * s_bwd[r][tid] + s_modes[r][1] * s_fwd[r][tid]
              + s_modes[r][2] * ccw;
    s_rwnew[r][tid] = rwn;
    rw[(b*R_ + r)*M_ + tid] = rwn;
  }
  __syncthreads();

  // ---- read vectors: last_read[r*64+w] = sum_m rw_new[r][m] * mem[m][w] ----
  for (int p = tid; p < R_ * W_; p += M_) {
    int r = p / W_, w = p - r * W_;
    float acc = 0.0f;
    for (int m = 0; m < M_; ++m) acc += s_rwnew[r][m] * s_mem[m][w];
    lread[b * (R_*W_) + p] = acc;
  }
}

extern "C" void kernel_launch(void* const* d_in, const int* in_sizes, int n_in,
                              void* d_out, int out_size, void* d_ws, size_t ws_size,
                              hipStream_t stream) {
  const float* x    = (const float*)d_in[0];
  const float* W_ih = (const float*)d_in[1];
  const float* W_hh = (const float*)d_in[2];
  const float* b_ih = (const float*)d_in[3];
  const float* b_hh = (const float*)d_in[4];
  const float* W_xi = (const float*)d_in[5];
  const float* b_xi = (const float*)d_in[6];
  const float* W_o  = (const float*)d_in[7];
  const float* b_o  = (const float*)d_in[8];
  float* out = (float*)d_out;
  float* ws  = (float*)d_ws;

  float* h     = ws + OFF_H;
  float* c     = ws + OFF_C;
  float* mem   = ws + OFF_MEM;
  float* link  = ws + OFF_LINK;
  float* prec  = ws + OFF_PREC;
  float* rw    = ws + OFF_RW;
  float* wwb   = ws + OFF_WW;
  float* usage = ws + OFF_USAGE;
  float* lread = ws + OFF_LREAD;
  float* gates = ws + OFF_GATES;
  float* xiw   = ws + OFF_XI;

  init_state<<<256, 256, 0, stream>>>(ws);

  for (int t = 0; t < T_; ++t) {
    gates_gemm<<<16, 256, 0, stream>>>(x, lread, h, W_ih, W_hh, b_ih, b_hh,
                                       gates, t);
    lstm_cell<<<(B_*H_ + 255)/256, 256, 0, stream>>>(gates, h, c);
    xi_gemm<<<15, 64, 0, stream>>>(h, W_xi, b_xi, xiw);
    memory_step<<<B_, 128, 0, stream>>>(mem, link, prec, rw, wwb, usage,
                                        lread, xiw);
    out_gemm<<<4, 256, 0, stream>>>(h, lread, W_o, b_o, out, t);
  }
}